// DGCNN_38869454029688
// MI455X (gfx1250) — compile-verified
//
#include <hip/hip_runtime.h>

#define BB   4
#define NPTS 2048
#define KNN  20
#define PARTS 50
#define EPSB 1e-3f

typedef __attribute__((ext_vector_type(2))) float v2f;
typedef __attribute__((ext_vector_type(8))) float v8f;

// ---------------------------------------------------------------- utilities
__global__ void zero_kernel(float* __restrict__ p, int n) {
    int t = blockIdx.x * blockDim.x + threadIdx.x;
    if (t < n) p[t] = 0.f;
}

// xx[b*N+n] = sum_c x[b][c][n]^2   (x has batch stride bstride, channel stride NPTS)
__global__ void xx_kernel(const float* __restrict__ x, float* __restrict__ xx,
                          int C, int bstride) {
    int t = blockIdx.x * blockDim.x + threadIdx.x;
    if (t >= BB * NPTS) return;
    int b = t / NPTS, n = t % NPTS;
    const float* xb = x + (size_t)b * bstride;
    float s = 0.f;
    for (int c = 0; c < C; ++c) { float v = xb[c * NPTS + n]; s += v * v; }
    xx[t] = s;
}

// one block per (b,n): top-KNN of pd[m] = 2*<x_n,x_m> - xx_n - xx_m
__global__ __launch_bounds__(256) void knn_kernel(const float* __restrict__ x,
                                                  const float* __restrict__ xx,
                                                  int* __restrict__ idx,
                                                  int C, int bstride) {
    __shared__ float dist[NPTS];
    __shared__ float xn[64];
    __shared__ float rv[256];
    __shared__ int   ri[256];
    int row = blockIdx.x;             // b*NPTS + n
    int b = row / NPTS, n = row % NPTS;
    int tid = threadIdx.x;
    const float* xb = x + (size_t)b * bstride;
    if (tid < C) xn[tid] = xb[tid * NPTS + n];
    __syncthreads();
    float xxn = xx[row];
    for (int m = tid; m < NPTS; m += 256) {
        float inner = 0.f;
        for (int c = 0; c < C; ++c) inner += xn[c] * xb[c * NPTS + m];
        dist[m] = 2.f * inner - xxn - xx[b * NPTS + m];
    }
    __syncthreads();
    for (int it = 0; it < KNN; ++it) {
        float bv = -3.4e38f; int bi = 0x7fffffff;
        for (int m = tid; m < NPTS; m += 256) {
            float d = dist[m];
            if (d > bv || (d == bv && m < bi)) { bv = d; bi = m; }
        }
        rv[tid] = bv; ri[tid] = bi;
        __syncthreads();
        for (int s = 128; s > 0; s >>= 1) {
            if (tid < s) {
                if (rv[tid + s] > rv[tid] ||
                    (rv[tid + s] == rv[tid] && ri[tid + s] < ri[tid])) {
                    rv[tid] = rv[tid + s]; ri[tid] = ri[tid + s];
                }
            }
            __syncthreads();
        }
        if (tid == 0) {
            idx[(size_t)row * KNN + it] = ri[0];
            dist[ri[0]] = -3.4e38f;
        }
        __syncthreads();
    }
}

// A[(b*N+n)*K + k][c] : c<C -> x[c][idx]-x[c][n]; c<2C -> x[c-C][n]; else 0
__global__ void edge_kernel(const float* __restrict__ x, const int* __restrict__ idx,
                            float* __restrict__ A, int C, int Cpad, int bstride) {
    int t = blockIdx.x * blockDim.x + threadIdx.x;
    int total = BB * NPTS * KNN * Cpad;
    if (t >= total) return;
    int r = t / Cpad, c = t % Cpad;
    int b = r / (NPTS * KNN);
    int n = (r / KNN) % NPTS;
    float val = 0.f;
    if (c < 2 * C) {
        const float* xb = x + (size_t)b * bstride;
        if (c < C) {
            int j = idx[r];
            val = xb[c * NPTS + j] - xb[c * NPTS + n];
        } else {
            val = xb[(c - C) * NPTS + n];
        }
    }
    A[t] = val;
}

// zero-padded weight copy:  Wp[o][c] (Opad x Cpad) from W[o][c] (O x C)
__global__ void padw_kernel(const float* __restrict__ W, float* __restrict__ Wp,
                            int O, int C, int Cpad, int Opad) {
    int t = blockIdx.x * blockDim.x + threadIdx.x;
    if (t >= Opad * Cpad) return;
    int o = t / Cpad, c = t % Cpad;
    Wp[t] = (o < O && c < C) ? W[o * C + c] : 0.f;
}

// fold conv-bias + batchnorm into alpha/beta:  y = acc*alpha + beta
__global__ void fold_kernel(const float* g, const float* bt, const float* m,
                            const float* v, const float* bias,
                            float* __restrict__ alpha, float* __restrict__ beta,
                            int O, int has_bn) {
    int o = blockIdx.x * blockDim.x + threadIdx.x;
    if (o >= O) return;
    float bi = bias ? bias[o] : 0.f;
    if (has_bn) {
        float s = g[o] * rsqrtf(v[o] + EPSB);
        alpha[o] = s;
        beta[o]  = (bi - m[o]) * s + bt[o];
    } else {
        alpha[o] = 1.f;
        beta[o]  = bi;
    }
}

// ------------------------------------------------- WMMA fp32 GEMM + epilogue
// D[m][o] = act( (A x W^T)*alpha + beta ).  A:[M][Cpad] rows, Wp:[Opad][Cpad].
// mode 0: store to out with leading dim ldo.
// mode 1: tw3 fused epilogue: atomic-max into out10240[b][o*10 + k/2], k even.
__global__ __launch_bounds__(128) void gemm_kernel(
        const float* __restrict__ A, const float* __restrict__ Wp,
        const float* __restrict__ alpha, const float* __restrict__ beta,
        float* __restrict__ out, unsigned int* __restrict__ omax,
        int M, int Cpad, int O, int ldo, int relu, int mode) {
    int lane = threadIdx.x & 31;
    int wave = threadIdx.x >> 5;
    int tilesO = (O + 15) >> 4;
    int tilesM = M >> 4;
    int tile = blockIdx.x * 4 + wave;
    if (tile >= tilesM * tilesO) return;
    int tm = tile / tilesO;
    int to = tile - tm * tilesO;
    int kb = (lane >> 4) << 1;                       // K sub-offset: 0 or 2
    const float* arow = A  + (size_t)(tm * 16 + (lane & 15)) * Cpad + kb;
    const float* wrow = Wp + (size_t)(to * 16 + (lane & 15)) * Cpad + kb;
    v8f acc = {};
    for (int c = 0; c < Cpad; c += 4) {
        v2f av, bv;
        av.x = arow[c];  av.y = arow[c + 1];
        bv.x = wrow[c];  bv.y = wrow[c + 1];
        acc = __builtin_amdgcn_wmma_f32_16x16x4_f32(false, av, false, bv,
                                                    (short)0, acc, false, false);
    }
    int col = to * 16 + (lane & 15);
    if (col >= O) return;
    float al = alpha[col], be = beta[col];
    int rbase = tm * 16 + ((lane >> 4) << 3);
    for (int r = 0; r < 8; ++r) {
        float v = acc[r] * al + be;
        if (relu) v = fmaxf(v, 0.f);
        int rr = rbase + r;
        if (mode == 0) {
            out[(size_t)rr * ldo + col] = v;
        } else {
            int k = rr % KNN;
            if ((k & 1) == 0) {
                int b = rr / (KNN * NPTS);
                atomicMax(&omax[b * 10240 + col * 10 + (k >> 1)],
                          __float_as_uint(v));        // v >= 0 after relu
            }
        }
    }
}

// max over K: out[b][coff+o][n] = max_k rows[((b*N+n)*K+k)*ld + o]
__global__ void maxk_kernel(const float* __restrict__ rows, float* __restrict__ out,
                            int O, int ld, int coff, int Ctot) {
    int t = blockIdx.x * blockDim.x + threadIdx.x;
    if (t >= BB * NPTS * O) return;
    int o = t % O;
    int n = (t / O) % NPTS;
    int b = t / (O * NPTS);
    const float* p = rows + ((size_t)(b * NPTS + n) * KNN) * ld + o;
    float m = -3.4e38f;
    for (int k = 0; k < KNN; ++k) m = fmaxf(m, p[k * ld]);
    out[(size_t)b * Ctot * NPTS + (coff + o) * NPTS + n] = m;
}

// om[b][o] = max_n rows[(b*N+n)*64 + o]
__global__ void om_kernel(const float* __restrict__ rows, float* __restrict__ om) {
    int t = threadIdx.x;                 // 256 = BB*64
    int b = t >> 6, o = t & 63;
    float m = -3.4e38f;
    for (int n = 0; n < NPTS; ++n)
        m = fmaxf(m, rows[(size_t)(b * NPTS + n) * 64 + o]);
    om[t] = m;
}

// rowsP[(b*N+n)*192 + c] = cat[b][c][n]
__global__ void pack1_kernel(const float* __restrict__ cat, float* __restrict__ rowsP) {
    int t = blockIdx.x * blockDim.x + threadIdx.x;
    if (t >= BB * NPTS * 192) return;
    int r = t / 192, c = t % 192;
    int b = r / NPTS, n = r % NPTS;
    rowsP[t] = cat[(size_t)b * 192 * NPTS + c * NPTS + n];
}

// rowsP[(b*N+n)*256 + c] = c<64 ? om[b][c] : cat[b][c-64][n]
__global__ void pack2_kernel(const float* __restrict__ om, const float* __restrict__ cat,
                             float* __restrict__ rowsP) {
    int t = blockIdx.x * blockDim.x + threadIdx.x;
    if (t >= BB * NPTS * 256) return;
    int r = t / 256, c = t % 256;
    int b = r / NPTS, n = r % NPTS;
    rowsP[t] = (c < 64) ? om[b * 64 + c]
                        : cat[(size_t)b * 192 * NPTS + (c - 64) * NPTS + n];
}

// tiny FC: out[b][o] = act(dot(in[b],W[o])*alpha + beta)
__global__ void fc_kernel(const float* __restrict__ in, const float* __restrict__ W,
                          const float* __restrict__ alpha, const float* __restrict__ beta,
                          float* __restrict__ out, int Cin, int O, int relu) {
    int t = blockIdx.x * blockDim.x + threadIdx.x;
    if (t >= BB * O) return;
    int b = t / O, o = t % O;
    const float* ip = in + (size_t)b * Cin;
    const float* wp = W + (size_t)o * Cin;
    float acc = 0.f;
    for (int c = 0; c < Cin; ++c) acc += ip[c] * wp[c];
    acc = acc * alpha[o] + beta[o];
    if (relu) acc = fmaxf(acc, 0.f);
    out[t] = acc;
}

// 3x3 transform head: T[b][i][j] = dot(in[b],fw3[o]) + fb3[o] + eye
__global__ void fct_kernel(const float* __restrict__ in, const float* __restrict__ W,
                           const float* __restrict__ bias, float* __restrict__ T) {
    int t = threadIdx.x;
    if (t >= BB * 9) return;
    int b = t / 9, o = t % 9;
    float acc = bias[o];
    for (int c = 0; c < 256; ++c) acc += in[b * 256 + c] * W[o * 256 + c];
    if (o == 0 || o == 4 || o == 8) acc += 1.f;
    T[t] = acc;
}

// xi[b][j][n] = sum_c x[b][c][n] * T[b][c*3+j]
__global__ void xform_kernel(const float* __restrict__ x, const float* __restrict__ T,
                             float* __restrict__ xi) {
    int t = blockIdx.x * blockDim.x + threadIdx.x;
    if (t >= BB * 3 * NPTS) return;
    int b = t / (3 * NPTS);
    int j = (t / NPTS) % 3;
    int n = t % NPTS;
    const float* xb = x + (size_t)b * 3 * NPTS;
    float s = 0.f;
    for (int c = 0; c < 3; ++c) s += xb[c * NPTS + n] * T[b * 9 + c * 3 + j];
    xi[t] = s;
}

// ---------------------------------------------------------------- launcher
extern "C" void kernel_launch(void* const* d_in, const int* in_sizes, int n_in,
                              void* d_out, int out_size, void* d_ws, size_t ws_size,
                              hipStream_t stream) {
    (void)in_sizes; (void)n_in; (void)out_size; (void)ws_size;
    const float* x = (const float*)d_in[0];
    // bn tuples flatten as (g, bt, m, v)
    const float *bn1g=(const float*)d_in[1],  *bn1b=(const float*)d_in[2],
                *bn1m=(const float*)d_in[3],  *bn1v=(const float*)d_in[4];
    const float *bn2g=(const float*)d_in[5],  *bn2b=(const float*)d_in[6],
                *bn2m=(const float*)d_in[7],  *bn2v=(const float*)d_in[8];
    const float *bn3g=(const float*)d_in[9],  *bn3b=(const float*)d_in[10],
                *bn3m=(const float*)d_in[11], *bn3v=(const float*)d_in[12];
    const float *bn5g=(const float*)d_in[13], *bn5b=(const float*)d_in[14],
                *bn5m=(const float*)d_in[15], *bn5v=(const float*)d_in[16];
    const float *bnfg=(const float*)d_in[17], *bnfb=(const float*)d_in[18],
                *bnfm=(const float*)d_in[19], *bnfv=(const float*)d_in[20];
    const float *bnf2g=(const float*)d_in[21],*bnf2b=(const float*)d_in[22],
                *bnf2m=(const float*)d_in[23],*bnf2v=(const float*)d_in[24];
    const float *tw1=(const float*)d_in[25], *tb1=(const float*)d_in[26];
    const float *tw2=(const float*)d_in[27], *tb2=(const float*)d_in[28];
    const float *tw3=(const float*)d_in[29], *tb3=(const float*)d_in[30];
    const float *fw1=(const float*)d_in[31], *fb1=(const float*)d_in[32];
    const float *fw2=(const float*)d_in[33], *fb2=(const float*)d_in[34];
    const float *fw3=(const float*)d_in[35], *fb3=(const float*)d_in[36];
    const float *w1=(const float*)d_in[37], *b1=(const float*)d_in[38];
    const float *w2=(const float*)d_in[39], *b2=(const float*)d_in[40];
    const float *w3=(const float*)d_in[41], *b3=(const float*)d_in[42];
    const float *w4=(const float*)d_in[43], *b4=(const float*)d_in[44];
    const float *w5=(const float*)d_in[45], *b5=(const float*)d_in[46];
    const float *w6=(const float*)d_in[47], *b6=(const float*)d_in[48];
    const float *w7=(const float*)d_in[49], *b7=(const float*)d_in[50];
    const float *w8=(const float*)d_in[51], *b8=(const float*)d_in[52];
    const float *w9=(const float*)d_in[53], *b9=(const float*)d_in[54];
    const float *w10=(const float*)d_in[55],*b10=(const float*)d_in[56];

    const int Me = BB * NPTS * KNN;   // 163840 edge rows
    const int Mp = BB * NPTS;         // 8192 point rows

    // workspace carve-out (floats)
    float* ws = (float*)d_ws;
    size_t off = 0;
    auto alloc = [&](size_t n) { float* p = ws + off; off += (n + 63) & ~(size_t)63; return p; };
    float* bufA  = alloc((size_t)Me * 128);
    float* bufB  = alloc((size_t)Me * 128);
    int*   idxb  = (int*)alloc((size_t)Me);
    float* xxb   = alloc(Mp);
    float* xib   = alloc((size_t)BB * 3 * NPTS);
    float* catb  = alloc((size_t)BB * 192 * NPTS);
    float* rowsP = alloc((size_t)Mp * 256);
    float* sbuf  = alloc((size_t)Mp * 256);
    float* omb   = alloc(BB * 64);
    float* o10k  = alloc(BB * 10240);
    float* fc1b  = alloc(BB * 512);
    float* fc2b  = alloc(BB * 256);
    float* Tmb   = alloc(BB * 9);
    float* wpad  = alloc(1024 * 128);
    float* alp   = alloc(1024);
    float* bet   = alloc(1024);

    auto cdiv = [](int a, int b) { return (a + b - 1) / b; };

    auto run_knn = [&](const float* xp, int C, int bstride) {
        xx_kernel<<<cdiv(BB * NPTS, 256), 256, 0, stream>>>(xp, xxb, C, bstride);
        knn_kernel<<<BB * NPTS, 256, 0, stream>>>(xp, xxb, idxb, C, bstride);
    };
    auto run_conv = [&](const float* Ain, int M, int C, int Cpad,
                        const float* W, const float* bias,
                        const float* g, const float* bt2, const float* mm, const float* vv,
                        int O, int ldo, int relu, float* out, int mode) {
        int Opad = cdiv(O, 16) * 16;
        padw_kernel<<<cdiv(Opad * Cpad, 256), 256, 0, stream>>>(W, wpad, O, C, Cpad, Opad);
        fold_kernel<<<cdiv(O, 256), 256, 0, stream>>>(g, bt2, mm, vv, bias, alp, bet, O,
                                                      g != nullptr);
        int tiles = (M / 16) * cdiv(O, 16);
        gemm_kernel<<<cdiv(tiles, 4), 128, 0, stream>>>(Ain, wpad, alp, bet, out,
                                                        (unsigned int*)o10k,
                                                        M, Cpad, O, ldo, relu, mode);
    };

    // -------- T-Net --------
    run_knn(x, 3, 3 * NPTS);
    edge_kernel<<<cdiv(Me * 8, 256), 256, 0, stream>>>(x, idxb, bufA, 3, 8, 3 * NPTS);
    run_conv(bufA, Me, 6, 8, tw1, tb1, bn1g, bn1b, bn1m, bn1v, 64, 64, 1, bufB, 0);
    run_conv(bufB, Me, 64, 64, tw2, tb2, bn2g, bn2b, bn2m, bn2v, 128, 128, 1, bufA, 0);
    zero_kernel<<<cdiv(BB * 10240, 256), 256, 0, stream>>>(o10k, BB * 10240);
    run_conv(bufA, Me, 128, 128, tw3, tb3, bn5g, bn5b, bn5m, bn5v, 1024, 0, 1, nullptr, 1);
    fold_kernel<<<cdiv(512, 256), 256, 0, stream>>>(bnf2g, bnf2b, bnf2m, bnf2v, fb1,
                                                    alp, bet, 512, 1);
    fc_kernel<<<cdiv(BB * 512, 256), 256, 0, stream>>>(o10k, fw1, alp, bet, fc1b, 10240, 512, 1);
    fold_kernel<<<1, 256, 0, stream>>>(bnfg, bnfb, bnfm, bnfv, fb2, alp, bet, 256, 1);
    fc_kernel<<<cdiv(BB * 256, 256), 256, 0, stream>>>(fc1b, fw2, alp, bet, fc2b, 512, 256, 1);
    fct_kernel<<<1, 64, 0, stream>>>(fc2b, fw3, fb3, Tmb);
    xform_kernel<<<cdiv(BB * 3 * NPTS, 256), 256, 0, stream>>>(x, Tmb, xib);

    // -------- DGCNN trunk --------
    run_knn(xib, 3, 3 * NPTS);
    edge_kernel<<<cdiv(Me * 8, 256), 256, 0, stream>>>(xib, idxb, bufA, 3, 8, 3 * NPTS);
    run_conv(bufA, Me, 6, 8, w1, b1, bn1g, bn1b, bn1m, bn1v, 64, 64, 1, bufB, 0);
    run_conv(bufB, Me, 64, 64, w2, b2, bn1g, bn1b, bn1m, bn1v, 64, 64, 1, bufA, 0);
    maxk_kernel<<<cdiv(Mp * 64, 256), 256, 0, stream>>>(bufA, catb, 64, 64, 0, 192);   // x1

    run_knn(catb, 64, 192 * NPTS);
    edge_kernel<<<cdiv(Me * 128, 256), 256, 0, stream>>>(catb, idxb, bufA, 64, 128, 192 * NPTS);
    run_conv(bufA, Me, 128, 128, w3, b3, bn1g, bn1b, bn1m, bn1v, 64, 64, 1, bufB, 0);
    run_conv(bufB, Me, 64, 64, w4, b4, bn1g, bn1b, bn1m, bn1v, 64, 64, 1, bufA, 0);
    maxk_kernel<<<cdiv(Mp * 64, 256), 256, 0, stream>>>(bufA, catb, 64, 64, 64, 192);  // x2

    run_knn(catb + 64 * NPTS, 64, 192 * NPTS);
    edge_kernel<<<cdiv(Me * 128, 256), 256, 0, stream>>>(catb + 64 * NPTS, idxb, bufA,
                                                         64, 128, 192 * NPTS);
    run_conv(bufA, Me, 128, 128, w5, b5, bn1g, bn1b, bn1m, bn1v, 64, 64, 1, bufB, 0);
    maxk_kernel<<<cdiv(Mp * 64, 256), 256, 0, stream>>>(bufB, catb, 64, 64, 128, 192); // x3

    // -------- head --------
    pack1_kernel<<<cdiv(Mp * 192, 256), 256, 0, stream>>>(catb, rowsP);
    run_conv(rowsP, Mp, 192, 192, w6, b6, bn1g, bn1b, bn1m, bn1v, 64, 64, 1, sbuf, 0);
    om_kernel<<<1, 256, 0, stream>>>(sbuf, omb);
    pack2_kernel<<<cdiv(Mp * 256, 256), 256, 0, stream>>>(omb, catb, rowsP);
    run_conv(rowsP, Mp, 256, 256, w7, b7, bn3g, bn3b, bn3m, bn3v, 256, 256, 1, sbuf, 0);
    run_conv(sbuf,  Mp, 256, 256, w8, b8, bn3g, bn3b, bn3m, bn3v, 256, 256, 1, rowsP, 0);
    run_conv(rowsP, Mp, 256, 256, w9, b9, bn2g, bn2b, bn2m, bn2v, 128, 128, 1, sbuf, 0);
    run_conv(sbuf,  Mp, 128, 128, w10, b10, nullptr, nullptr, nullptr, nullptr,
             PARTS, PARTS, 1, (float*)d_out, 0);
}